// Loss_Compute_11269994185052
// MI455X (gfx1250) — compile-verified
//
#include <hip/hip_runtime.h>
#include <cmath>

#define LEN_VOCAB 50257
#define BATCH     2048
#define K_LABELS  5
#define NBLOCKS   4096
#define NTHREADS  256

typedef float v2f __attribute__((ext_vector_type(2)));
typedef float v4f __attribute__((ext_vector_type(4)));
typedef float v8f __attribute__((ext_vector_type(8)));

// Wave-level (32-lane) sum using the CDNA5 matrix unit.
// A = 16x4 f32 holding the 32 per-lane partials (lane L<16 -> A[L,0], lane L>=16 -> A[L-16,2]),
// B = 4x16 ones  =>  D[m,n] = s_m + s_{m+16}. Summing the 8 D VGPRs per lane gives the
// rows-0..7 total in lanes 0-15 and rows-8..15 total in lanes 16-31; one xor-16 shuffle
// combines the halves. Full f32 precision, one v_wmma instead of a 5-step shuffle tree.
__device__ __forceinline__ float wave_reduce(float s) {
#if __has_builtin(__builtin_amdgcn_wmma_f32_16x16x4_f32)
    v2f a; a.x = s;    a.y = 0.0f;
    v2f b; b.x = 1.0f; b.y = 1.0f;
    v8f c = {};
    c = __builtin_amdgcn_wmma_f32_16x16x4_f32(false, a, false, b, (short)0, c, false, false);
    float t = ((c[0] + c[1]) + (c[2] + c[3])) + ((c[4] + c[5]) + (c[6] + c[7]));
    t += __shfl_xor(t, 16);
    return t;
#else
    for (int off = 16; off > 0; off >>= 1) s += __shfl_xor(s, off);
    return s;
#endif
}

// Kernel 1: streaming partial sums (+ per-row label term for blocks < BATCH).
// partials[blk] = -fill * (sum of this block's chunk of `output`)
//                 [+ V*fill*log(fill) + U_b*(lv*log(lv)-fill*log(fill)) + (fill-lv)*L_b]
__global__ __launch_bounds__(NTHREADS) void
kl_partial_kernel(const float* __restrict__ out_logq,
                  const int*   __restrict__ labels,
                  float*       __restrict__ partials,
                  long long nvec4, float fill,
                  float c_base, float c_uniq, float c_lab)
{
    const v4f* __restrict__ p4 = (const v4f*)out_logq;

    v4f acc; acc.x = 0.0f; acc.y = 0.0f; acc.z = 0.0f; acc.w = 0.0f;
    const long long stride = (long long)gridDim.x * NTHREADS;
    for (long long i = (long long)blockIdx.x * NTHREADS + threadIdx.x; i < nvec4; i += stride) {
        // Non-temporal: 412MB stream > 192MB L2; don't thrash the cache.
        v4f v = __builtin_nontemporal_load(&p4[i]);
        acc.x += v.x; acc.y += v.y; acc.z += v.z; acc.w += v.w;
    }
    float s = (acc.x + acc.y) + (acc.z + acc.w);

    // Straight-line, all 256 lanes active -> EXEC all-ones for the WMMA.
    float wt = wave_reduce(s);

    __shared__ float wsum[NTHREADS / 32];
    const int lane = threadIdx.x & 31;
    const int wid  = threadIdx.x >> 5;
    if (lane == 0) wsum[wid] = wt;
    __syncthreads();

    if (threadIdx.x == 0) {
        float tot = 0.0f;
        #pragma unroll
        for (int w = 0; w < NTHREADS / 32; ++w) tot += wsum[w];

        float res = -fill * tot;

        const int row = blockIdx.x;
        if (row < BATCH) {
            int   labs[K_LABELS];
            float L = 0.0f;
            int   u = 0;
            #pragma unroll
            for (int k = 0; k < K_LABELS; ++k) {
                const int l = labels[row * K_LABELS + k];
                bool dup = false;
                #pragma unroll
                for (int j = 0; j < K_LABELS; ++j)
                    if (j < k) dup |= (labs[j] == l);
                labs[k] = l;
                if (!dup) {
                    ++u;
                    L += out_logq[(long long)row * LEN_VOCAB + l];
                }
            }
            res += c_base + (float)u * c_uniq + c_lab * L;
        }
        partials[row] = res;
    }
}

// Kernel 2: reduce the per-block partials and scale by 1/B.
__global__ __launch_bounds__(NTHREADS) void
kl_finalize_kernel(const float* __restrict__ partials, int np,
                   float* __restrict__ out, float invB)
{
    float s = 0.0f;
    for (int i = threadIdx.x; i < np; i += NTHREADS) s += partials[i];

    float wt = wave_reduce(s);

    __shared__ float wsum[NTHREADS / 32];
    if ((threadIdx.x & 31) == 0) wsum[threadIdx.x >> 5] = wt;
    __syncthreads();

    if (threadIdx.x == 0) {
        float t = 0.0f;
        #pragma unroll
        for (int w = 0; w < NTHREADS / 32; ++w) t += wsum[w];
        out[0] = t * invB;
    }
}

extern "C" void kernel_launch(void* const* d_in, const int* in_sizes, int n_in,
                              void* d_out, int out_size, void* d_ws, size_t ws_size,
                              hipStream_t stream)
{
    (void)in_sizes; (void)n_in; (void)out_size; (void)ws_size;

    const float* out_logq = (const float*)d_in[0];
    const int*   labels   = (const int*)d_in[1];
    float*       loss     = (float*)d_out;
    float*       partials = (float*)d_ws;   // NBLOCKS floats, fully rewritten each call

    const double V    = (double)LEN_VOCAB;
    const double K    = (double)K_LABELS;
    const double fill = 0.1 / (V - K);          // mass on non-label entries
    const double lv   = (1.0 - 0.1) / K;        // mass per label entry

    const float c_base = (float)(V * fill * std::log(fill));
    const float c_uniq = (float)(lv * std::log(lv) - fill * std::log(fill));
    const float c_lab  = (float)(fill - lv);

    const long long nvec4 = ((long long)BATCH * LEN_VOCAB) / 4; // divisible: 2048 % 4 == 0

    kl_partial_kernel<<<NBLOCKS, NTHREADS, 0, stream>>>(
        out_logq, labels, partials, nvec4, (float)fill, c_base, c_uniq, c_lab);

    kl_finalize_kernel<<<1, NTHREADS, 0, stream>>>(
        partials, NBLOCKS, loss, (float)(1.0 / (double)BATCH));
}